// KalmanFilter_7310034338084
// MI455X (gfx1250) — compile-verified
//
#include <hip/hip_runtime.h>
#include <hip/hip_bf16.h>

typedef __attribute__((ext_vector_type(2))) float v2f;
typedef __attribute__((ext_vector_type(8))) float v8f;

#define DEV __device__ __forceinline__

namespace {

constexpr int B_ = 256, T_ = 128, DZ = 32, DA = 16, KM = 8, DH = 64, DG = 64;
constexpr long long BT = (long long)B_ * T_;

// output offsets (in floats), in reference return order
constexpr long long OFF_ZF  = 0;                       // z_filt      [B,T,32]
constexpr long long OFF_PF  = OFF_ZF  + BT * DZ;       // P_filt      [B,T,32,32]
constexpr long long OFF_ZP  = OFF_PF  + BT * DZ * DZ;  // z_pred      [B,T,32]
constexpr long long OFF_AF  = OFF_ZP  + BT * DZ;       // a_filt      [B,T,16]
constexpr long long OFF_AP  = OFF_AF  + BT * DA;       // a_pred      [B,T,16]
constexpr long long OFF_PP  = OFF_AP  + BT * DA;       // P_pred      [B,T,32,32]
constexpr long long OFF_AL  = OFF_PP  + BT * DZ * DZ;  // alpha_seq   [B,T,8]
constexpr long long OFF_AML = OFF_AL  + BT * KM;       // A_list      [B,T,32,32]
constexpr long long OFF_CML = OFF_AML + BT * DZ * DZ;  // C_list      [B,T,16,32]
constexpr long long OFF_ZM  = OFF_CML + BT * DA * DZ;  // z_means     [B,T,32]
constexpr long long OFF_LT  = OFF_ZM  + BT * DZ;       // z_scale_tril[B,T,32,32]
constexpr long long OFF_SP  = OFF_LT  + BT * DZ * DZ;  // S_pred      [B,T,16,16]

DEV float sigm(float x) { return 1.0f / (1.0f + __expf(-x)); }

// D[MxN] = opA(A)[MxK] * opB(B)[KxN], all row-major f32 in LDS.
// TA: A argument stores A^T (A[m][k] = Astore[k][m]); TB analogous.
// Uses V_WMMA_F32_16X16X4_F32 per the CDNA5 VGPR layouts:
//   A-frag: lane m = lane&15; vgpr0 = K + 2*(lane>>4), vgpr1 = +1
//   B-frag: lane n = lane&15; vgpr0 = row K + 2*(lane>>4), vgpr1 = +1
//   C/D:    col = lane&15; vgpr r -> row r + 8*(lane>>4)
template <int M, int N, int K, bool TA, bool TB>
DEV void wmma_gemm(float* D, int ldd, const float* A, int lda,
                   const float* Bm, int ldb, int lane) {
  const int lo = lane & 15, hi = lane >> 4;
#pragma unroll
  for (int r0 = 0; r0 < M; r0 += 16) {
#pragma unroll
    for (int n0 = 0; n0 < N; n0 += 16) {
      v8f acc = {};
#pragma unroll
      for (int k0 = 0; k0 < K; k0 += 4) {
        v2f af, bf;
        if (!TA) {
          af.x = A[(r0 + lo) * lda + (k0 + 2 * hi)];
          af.y = A[(r0 + lo) * lda + (k0 + 2 * hi + 1)];
        } else {
          af.x = A[(k0 + 2 * hi) * lda + (r0 + lo)];
          af.y = A[(k0 + 2 * hi + 1) * lda + (r0 + lo)];
        }
        if (!TB) {
          bf.x = Bm[(k0 + 2 * hi) * ldb + (n0 + lo)];
          bf.y = Bm[(k0 + 2 * hi + 1) * ldb + (n0 + lo)];
        } else {
          bf.x = Bm[(n0 + lo) * ldb + (k0 + 2 * hi)];
          bf.y = Bm[(n0 + lo) * ldb + (k0 + 2 * hi + 1)];
        }
        acc = __builtin_amdgcn_wmma_f32_16x16x4_f32(false, af, false, bf,
                                                    (short)0, acc, false, false);
      }
#pragma unroll
      for (int r = 0; r < 8; ++r)
        D[(r0 + r + 8 * hi) * ldd + (n0 + lo)] = acc[r];
    }
  }
}

// GRU cell + softmax(alpha). h is 64 floats in LDS, xv is 80 floats in LDS: a (16), then h_ctx (64).
DEV void gru_alpha(const float* __restrict__ Wx, const float* __restrict__ Wh,
                   const float* __restrict__ bg, const float* __restrict__ Wo,
                   const float* __restrict__ bo, const float* xv, float* h,
                   float* lg, float* al, int lane) {
  float gx[6], gh[6];
#pragma unroll
  for (int c = 0; c < 6; ++c) {
    const int j = c * 32 + lane;
    float g = bg[j];
    for (int i = 0; i < 80; ++i) g += xv[i] * Wx[i * 192 + j];
    gx[c] = g;
    float g2 = 0.f;
    for (int i = 0; i < 64; ++i) g2 += h[i] * Wh[i * 192 + j];
    gh[c] = g2;
  }
  const float h0 = h[lane], h1 = h[32 + lane];
  const float r0 = sigm(gx[0] + gh[0]), r1 = sigm(gx[1] + gh[1]);
  const float u0 = sigm(gx[2] + gh[2]), u1 = sigm(gx[3] + gh[3]);
  const float n0 = tanhf(gx[4] + r0 * gh[4]), n1 = tanhf(gx[5] + r1 * gh[5]);
  __syncthreads();
  h[lane]      = (1.f - u0) * n0 + u0 * h0;
  h[32 + lane] = (1.f - u1) * n1 + u1 * h1;
  __syncthreads();
  if (lane < 8) {
    float g = bo[lane];
    for (int i = 0; i < 64; ++i) g += h[i] * Wo[i * 8 + lane];
    lg[lane] = g;
  }
  __syncthreads();
  if (lane < 8) {
    float m = lg[0];
    for (int k = 1; k < 8; ++k) m = fmaxf(m, lg[k]);
    float s = 0.f;
    for (int k = 0; k < 8; ++k) s += __expf(lg[k] - m);
    al[lane] = __expf(lg[lane] - m) / s;
  }
  __syncthreads();
}

// A_mix = sum_k alpha_k A_k ; C_mix = sum_k alpha_k C_k (weights from L2/WGP$)
DEV void mix_mats(const float* __restrict__ Ag, const float* __restrict__ Cg,
                  const float* al, float* Am, float* Cm, int lane) {
  float a[8];
#pragma unroll
  for (int k = 0; k < 8; ++k) a[k] = al[k];
  for (int e = 0; e < 32; ++e) {
    float acc = 0.f;
#pragma unroll
    for (int k = 0; k < 8; ++k) acc += a[k] * Ag[k * 1024 + e * 32 + lane];
    Am[e * 32 + lane] = acc;
  }
  for (int e = 0; e < 16; ++e) {
    float acc = 0.f;
#pragma unroll
    for (int k = 0; k < 8; ++k) acc += a[k] * Cg[k * 512 + e * 32 + lane];
    Cm[e * 32 + lane] = acc;
  }
}

} // namespace

__global__ __launch_bounds__(32) void kf_mix_kernel(
    const float* __restrict__ a_seq, const float* __restrict__ h_obs,
    const float* __restrict__ Ag, const float* __restrict__ Cg,
    const float* __restrict__ a0, const float* __restrict__ Wx,
    const float* __restrict__ Wh, const float* __restrict__ bg,
    const float* __restrict__ Wo, const float* __restrict__ bo,
    float* __restrict__ out) {
  __shared__ float sP[1024], sPf[1024], sAm[1024], sIKC[1024], sTt[1024], sLb[1024];
  __shared__ float sCm[512], sX[512];
  __shared__ float sS[256];
  __shared__ float sh[64], shc[64], sxv[80];
  __shared__ float sz[32], szt[32], szf[32];
  __shared__ float sr[16];
  __shared__ float slg[8], sal[8];

  const int lane = threadIdx.x;
  const int b = blockIdx.x;

  // ---- init: h_ctx = mean_t h_obs[b], GRU state 0, z0=0, P0=10I ----
  {
    float s0 = 0.f, s1 = 0.f;
    const float* hb = h_obs + (size_t)b * T_ * DH;
    for (int t = 0; t < T_; ++t) {
      s0 += hb[t * DH + lane];
      s1 += hb[t * DH + 32 + lane];
    }
    shc[lane] = s0 * (1.0f / T_);
    shc[32 + lane] = s1 * (1.0f / T_);
    sh[lane] = 0.f;
    sh[32 + lane] = 0.f;
  }
  __syncthreads();
  sxv[16 + lane] = shc[lane];
  sxv[48 + lane] = shc[32 + lane];
  if (lane < DA) sxv[lane] = a0[lane];
  sz[lane] = 0.f;
  for (int e = 0; e < 32; ++e) sP[e * 32 + lane] = (e == lane) ? 10.f : 0.f;
  __syncthreads();

  gru_alpha(Wx, Wh, bg, Wo, bo, sxv, sh, slg, sal, lane);  // alpha0 from a_0
  mix_mats(Ag, Cg, sal, sAm, sCm, lane);                   // A_k0, C_k0
  __syncthreads();

#pragma unroll 1
  for (int t = 0; t < T_; ++t) {
    const size_t bt = (size_t)b * T_ + (size_t)t;

    if (lane < DA) sxv[lane] = a_seq[bt * DA + lane];  // a_k
    __syncthreads();

    // z = A_k z_prev
    {
      float s = 0.f;
      for (int j = 0; j < 32; ++j) s += sAm[lane * 32 + j] * sz[j];
      szt[lane] = s;
    }
    __syncthreads();
    // r = a_k - C_k z
    if (lane < DA) {
      float ah = 0.f;
      for (int j = 0; j < 32; ++j) ah += sCm[lane * 32 + j] * szt[j];
      sr[lane] = sxv[lane] - ah;
    }
    __syncthreads();

    // Tt = C * P  (16x32)
    wmma_gemm<16, 32, 32, false, false>(sTt, 32, sCm, 32, sP, 32, lane);
    __syncthreads();
    // S = Tt * C^T + R
    wmma_gemm<16, 16, 32, false, true>(sS, 16, sTt, 32, sCm, 32, lane);
    __syncthreads();
    if (lane < 16) sS[lane * 16 + lane] += 0.3f;
    __syncthreads();
    {
      float* o = out + OFF_SP + (long long)bt * 256;
      for (int e = 0; e < 8; ++e) o[e * 32 + lane] = sS[e * 32 + lane];
    }
    __syncthreads();

    // chol(S) in place (lower, 16x16)
    for (int k = 0; k < 16; ++k) {
      const float dk = sqrtf(sS[k * 16 + k]);
      if (lane < 16) {
        float v = sS[lane * 16 + k];
        if (lane == k) v = dk;
        else if (lane > k) v = v / dk;
        sS[lane * 16 + k] = v;
      }
      __syncthreads();
      if (lane < 16 && lane > k) {
        const float lk = sS[lane * 16 + k];
        for (int j = k + 1; j <= lane; ++j) sS[lane * 16 + j] -= lk * sS[j * 16 + k];
      }
      __syncthreads();
    }

    // X = S^{-1} (C P): fwd/back substitution, one RHS column per lane
    {
      float y[16], xx[16];
#pragma unroll
      for (int i = 0; i < 16; ++i) {
        float s = sTt[i * 32 + lane];
#pragma unroll
        for (int j = 0; j < 16; ++j)
          if (j < i) s -= sS[i * 16 + j] * y[j];
        y[i] = s / sS[i * 16 + i];
      }
#pragma unroll
      for (int i = 15; i >= 0; --i) {
        float s = y[i];
#pragma unroll
        for (int j = 0; j < 16; ++j)
          if (j > i) s -= sS[j * 16 + i] * xx[j];
        xx[i] = s / sS[i * 16 + i];
      }
#pragma unroll
      for (int i = 0; i < 16; ++i) sX[i * 32 + lane] = xx[i];
    }
    __syncthreads();

    // KK = K K^T = X^T X   (for K R K^T term; R = 0.3 I)
    wmma_gemm<32, 32, 16, true, false>(sLb, 32, sX, 32, sX, 32, lane);
    // KC = K C = X^T C
    wmma_gemm<32, 32, 16, true, false>(sTt, 32, sX, 32, sCm, 32, lane);
    __syncthreads();
    for (int e = 0; e < 32; ++e)
      sIKC[e * 32 + lane] = ((e == lane) ? 1.f : 0.f) - sTt[e * 32 + lane];
    __syncthreads();
    // T2 = IKC * P
    wmma_gemm<32, 32, 32, false, false>(sTt, 32, sIKC, 32, sP, 32, lane);
    __syncthreads();
    // Pf = T2 * IKC^T
    wmma_gemm<32, 32, 32, false, true>(sPf, 32, sTt, 32, sIKC, 32, lane);
    __syncthreads();
    // Pf = 0.5*((Pf + 0.3 KK) + transpose)
    for (int e = 0; e < 32; ++e) {
      const float v1 = sPf[e * 32 + lane] + 0.3f * sLb[e * 32 + lane];
      const float v2 = sPf[lane * 32 + e] + 0.3f * sLb[lane * 32 + e];
      sTt[e * 32 + lane] = 0.5f * (v1 + v2);
    }
    __syncthreads();
    {
      float* o = out + OFF_PF + (long long)bt * 1024;
      for (int e = 0; e < 32; ++e) {
        const float v = sTt[e * 32 + lane];
        sPf[e * 32 + lane] = v;
        o[e * 32 + lane] = v;
      }
    }
    __syncthreads();

    // z_f = z + K r  (K[i][j] = X[j][i])
    {
      float s = szt[lane];
      for (int j = 0; j < 16; ++j) s += sX[j * 32 + lane] * sr[j];
      szf[lane] = s;
      sz[lane] = s;
      out[OFF_ZF + (long long)bt * 32 + lane] = s;
      out[OFF_ZM + (long long)bt * 32 + lane] = s;
    }
    __syncthreads();
    // a_filt = C_k(old) z_f ; C_list = C_k(old)
    if (lane < DA) {
      float s = 0.f;
      for (int j = 0; j < 32; ++j) s += sCm[lane * 32 + j] * szf[j];
      out[OFF_AF + (long long)bt * 16 + lane] = s;
    }
    {
      float* o = out + OFF_CML + (long long)bt * 512;
      for (int e = 0; e < 16; ++e) o[e * 32 + lane] = sCm[e * 32 + lane];
    }
    __syncthreads();

    // L = chol(Pf + 2*JITTER*I), lower 32x32 (row per lane)
    for (int e = 0; e < 32; ++e)
      sLb[e * 32 + lane] =
          (lane < e) ? sPf[e * 32 + lane]
                     : ((lane == e) ? sPf[e * 32 + lane] + 2.0e-4f : 0.f);
    __syncthreads();
    for (int k = 0; k < 32; ++k) {
      const float dk = sqrtf(sLb[k * 32 + k]);
      float v = sLb[lane * 32 + k];
      if (lane == k) v = dk;
      else if (lane > k) v = v / dk;
      sLb[lane * 32 + k] = v;
      __syncthreads();
      if (lane > k) {
        for (int j = k + 1; j <= lane; ++j) sLb[lane * 32 + j] -= v * sLb[j * 32 + k];
      }
      __syncthreads();
    }
    {
      float* o = out + OFF_LT + (long long)bt * 1024;
      for (int e = 0; e < 32; ++e) o[e * 32 + lane] = sLb[e * 32 + lane];
    }
    __syncthreads();

    // alpha step on true observation, then new mixture
    gru_alpha(Wx, Wh, bg, Wo, bo, sxv, sh, slg, sal, lane);
    if (lane < 8) out[OFF_AL + (long long)bt * 8 + lane] = sal[lane];
    mix_mats(Ag, Cg, sal, sAm, sCm, lane);
    __syncthreads();
    {
      float* o = out + OFF_AML + (long long)bt * 1024;
      for (int e = 0; e < 32; ++e) o[e * 32 + lane] = sAm[e * 32 + lane];
    }
    // z_pred = A2 z_f ; a_pred = C2 z_pred
    {
      float s = 0.f;
      for (int j = 0; j < 32; ++j) s += sAm[lane * 32 + j] * sz[j];
      szt[lane] = s;
      out[OFF_ZP + (long long)bt * 32 + lane] = s;
    }
    __syncthreads();
    if (lane < DA) {
      float s = 0.f;
      for (int j = 0; j < 32; ++j) s += sCm[lane * 32 + j] * szt[j];
      out[OFF_AP + (long long)bt * 16 + lane] = s;
    }
    __syncthreads();

    // P_pred = A2 Pf A2^T + Q, symmetrized -> carry P, output
    wmma_gemm<32, 32, 32, false, false>(sTt, 32, sAm, 32, sPf, 32, lane);
    __syncthreads();
    wmma_gemm<32, 32, 32, false, true>(sP, 32, sTt, 32, sAm, 32, lane);
    __syncthreads();
    for (int e = 0; e < 32; ++e)
      sTt[e * 32 + lane] = 0.5f * (sP[e * 32 + lane] + sP[lane * 32 + e]) +
                           ((e == lane) ? 0.2f : 0.f);
    __syncthreads();
    {
      float* o = out + OFF_PP + (long long)bt * 1024;
      for (int e = 0; e < 32; ++e) {
        const float v = sTt[e * 32 + lane];
        sP[e * 32 + lane] = v;
        o[e * 32 + lane] = v;
      }
    }
    __syncthreads();
  }
}

extern "C" void kernel_launch(void* const* d_in, const int* in_sizes, int n_in,
                              void* d_out, int out_size, void* d_ws, size_t ws_size,
                              hipStream_t stream) {
  (void)in_sizes; (void)n_in; (void)d_ws; (void)ws_size; (void)out_size;
  const float* a_seq = (const float*)d_in[0];
  const float* h_obs = (const float*)d_in[1];
  const float* Ag    = (const float*)d_in[2];
  const float* Cg    = (const float*)d_in[3];
  const float* a0    = (const float*)d_in[4];
  const float* Wx    = (const float*)d_in[5];
  const float* Wh    = (const float*)d_in[6];
  const float* bg    = (const float*)d_in[7];
  const float* Wo    = (const float*)d_in[8];
  const float* bo    = (const float*)d_in[9];
  float* out = (float*)d_out;
  kf_mix_kernel<<<dim3(B_), dim3(32), 0, stream>>>(a_seq, h_obs, Ag, Cg, a0, Wx,
                                                   Wh, bg, Wo, bo, out);
}